// DynSCFGaussian_31765578121909
// MI455X (gfx1250) — compile-verified
//
#include <hip/hip_runtime.h>
#include <math.h>

// ---------------------------------------------------------------------------
// DynSCF Gaussian skinning for MI455X (gfx1250, wave32).
//
// N=200000 gaussians, M=10000 nodes, K=16 neighbors, T=40 frames, F=32 feats.
// Outputs flat: mu(N,3) | fr(N,3,3) | s(N,3) | o(N,1) | sph(N,12) | sem(N,32).
//
// Workload analysis: ~0.8 GFLOP fp32, ~70 MB streaming HBM, ~600 MB of random
// 16-128B gathers into L2-resident node tables, ~10M transcendentals.
// -> L2-gather-latency + VALU bound. WMMA inapplicable (all blends are
// block-diagonal per gaussian; no shared operand). CDNA5 paths used instead:
// async global->LDS staging (ASYNCcnt), global_prefetch_b8, v_rsq/v_rcp/v_exp
// fast transcendentals, wave32-sized blocks, b128 vector memory ops.
// ---------------------------------------------------------------------------

namespace {

constexpr int   KNN   = 16;
constexpr int   FEAT  = 32;
constexpr int   M_MAX = 10000;    // LDS sigma-table capacity (M == 10000 here)
constexpr float MAX_S = 0.1f;
constexpr float EPS   = 1e-8f;
constexpr int   BLOCK = 256;

// ---------------- CDNA5 async global->LDS path (probe-verified) ------------
#if defined(__HIP_DEVICE_COMPILE__)
#  if __has_builtin(__builtin_amdgcn_global_load_async_to_lds_b128)
#    define HAVE_ASYNC_G2L 1
#  else
#    define HAVE_ASYNC_G2L 0
#  endif
#else
#  define HAVE_ASYNC_G2L 0
#endif

// Builtin takes pointers to GCC-vector int4 (confirmed by clang diagnostic).
typedef int v4i_t __attribute__((vector_size(16)));

__device__ __forceinline__ void g2l_async_b128(const float* g, float* l) {
#if HAVE_ASYNC_G2L
  // emits global_load_async_to_lds_b128 (tracked by ASYNCcnt)
  __builtin_amdgcn_global_load_async_to_lds_b128((v4i_t*)g, (v4i_t*)l,
                                                 /*offset=*/0, /*cpol=*/0);
#else
  *(float4*)l = *(const float4*)g;
#endif
}

__device__ __forceinline__ void wait_async_zero() {
#if HAVE_ASYNC_G2L
#  if __has_builtin(__builtin_amdgcn_s_wait_asynccnt)
  __builtin_amdgcn_s_wait_asynccnt(0);
#  else
  asm volatile("s_wait_asynccnt 0x0" ::: "memory");
#  endif
#endif
}

// ---------------- fast math helpers (single-instruction HW paths) ----------
__device__ __forceinline__ float fast_rcp(float x) {
  return __builtin_amdgcn_rcpf(x);          // v_rcp_f32
}
__device__ __forceinline__ float fast_sigmoid(float x) {
  return fast_rcp(1.0f + __expf(-x));       // v_exp_f32 + v_rcp_f32
}
// 1 / clip(sqrt(n2), 1e-12): rsqrt(max(n2,1e-24)) is identical (rsq(1e-24)=1e12)
__device__ __forceinline__ float rnorm_clip(float n2) {
  return __builtin_amdgcn_rsqf(fmaxf(n2, 1e-24f));   // v_rsq_f32
}

// q -> 3x3 rotation, expects (w,x,y,z) already unit-length
__device__ __forceinline__ void quat_to_R(float w, float x, float y, float z,
                                          float* R) {
  const float xx = x * x, yy = y * y, zz = z * z;
  const float xy = x * y, xz = x * z, yz = y * z;
  const float wx = w * x, wy = w * y, wz = w * z;
  R[0] = 1.0f - 2.0f * (yy + zz); R[1] = 2.0f * (xy - wz); R[2] = 2.0f * (xz + wy);
  R[3] = 2.0f * (xy + wz); R[4] = 1.0f - 2.0f * (xx + zz); R[5] = 2.0f * (yz - wx);
  R[6] = 2.0f * (xz - wy); R[7] = 2.0f * (yz + wx); R[8] = 1.0f - 2.0f * (xx + yy);
}

__device__ __forceinline__ void mat3mul(const float* A, const float* B, float* C) {
#pragma unroll
  for (int i = 0; i < 3; ++i)
#pragma unroll
    for (int j = 0; j < 3; ++j)
      C[3 * i + j] = A[3 * i + 0] * B[j] + A[3 * i + 1] * B[3 + j] +
                     A[3 * i + 2] * B[6 + j];
}

// ---------------- prep: pack live-frame node pose (amortized M << N) -------
// pack[2m]   = { p_live.x, p_live.y, p_live.z, q_live_n.w }
// pack[2m+1] = { q_live_n.x, q_live_n.y, q_live_n.z, 0 }
__global__ __launch_bounds__(BLOCK) void prep_nodes_kernel(
    const float* __restrict__ node_xyz, const float* __restrict__ node_quat,
    const int* __restrict__ tptr, float4* __restrict__ pack, int M) {
  const int m = blockIdx.x * BLOCK + threadIdx.x;
  if (m >= M) return;
  const int t = *tptr;
  const size_t b = (size_t)t * (size_t)M + (size_t)m;
  const float* p = node_xyz + 3 * b;
  const float4 q = *(const float4*)(node_quat + 4 * b);
  const float rn = rnorm_clip(q.x * q.x + q.y * q.y + q.z * q.z + q.w * q.w);
  pack[2 * (size_t)m]     = make_float4(p[0], p[1], p[2], q.x * rn);
  pack[2 * (size_t)m + 1] = make_float4(q.y * rn, q.z * rn, q.w * rn, 0.0f);
}

// ---------------- main kernel ----------------------------------------------
template <bool USE_PACK>
__global__ __launch_bounds__(BLOCK) void dynscf_kernel(
    const float* __restrict__ gs_xyz, const float* __restrict__ gs_rot,
    const float* __restrict__ gs_scal, const float* __restrict__ gs_opa,
    const float* __restrict__ feat_dc, const float* __restrict__ feat_rest,
    const float* __restrict__ node_xyz, const float* __restrict__ node_quat,
    const float* __restrict__ node_sigma, const float* __restrict__ node_sem,
    const int* __restrict__ attach, const int* __restrict__ reft,
    const int* __restrict__ knn, const int* __restrict__ tptr,
    const float4* __restrict__ pack, float* __restrict__ out, int N, int M) {
  // ---- stage 1/(2*sigma^2+eps) into LDS via CDNA5 async copies ------------
  __shared__ float s_inv[M_MAX];
  {
    const int Mc = (M < M_MAX) ? M : M_MAX;
    const int Mv = Mc & ~3;
    for (int i = threadIdx.x; 4 * i < Mv; i += BLOCK)
      g2l_async_b128(node_sigma + 4 * (size_t)i, s_inv + 4 * i);
    for (int i = Mv + threadIdx.x; i < Mc; i += BLOCK)  // tail (M%4), none here
      s_inv[i] = node_sigma[i];
    wait_async_zero();
    __syncthreads();
    for (int i = threadIdx.x; i < Mc; i += BLOCK) {
      const float sg = s_inv[i];
      s_inv[i] = fast_rcp(2.0f * sg * sg + EPS);
    }
    __syncthreads();
  }

  const int n = blockIdx.x * BLOCK + threadIdx.x;
  if (n >= N) return;

  const int a  = attach[n];
  const int rt = reft[n];
  const size_t refBase  = (size_t)rt * (size_t)M;
  const size_t liveBase = USE_PACK ? 0 : (size_t)(*tptr) * (size_t)M;

  // ---- attach node reference pose -> x_world, R_world ---------------------
  const float4 qa = *(const float4*)(node_quat + 4 * (refBase + (size_t)a));
  const float rna = rnorm_clip(qa.x * qa.x + qa.y * qa.y + qa.z * qa.z + qa.w * qa.w);
  float Rr[9];
  quat_to_R(qa.x * rna, qa.y * rna, qa.z * rna, qa.w * rna, Rr);

  const float* pa = node_xyz + 3 * (refBase + (size_t)a);
  const float gx = gs_xyz[3 * (size_t)n + 0];
  const float gy = gs_xyz[3 * (size_t)n + 1];
  const float gz = gs_xyz[3 * (size_t)n + 2];
  const float xwx = Rr[0] * gx + Rr[1] * gy + Rr[2] * gz + pa[0];
  const float xwy = Rr[3] * gx + Rr[4] * gy + Rr[5] * gz + pa[1];
  const float xwz = Rr[6] * gx + Rr[7] * gy + Rr[8] * gz + pa[2];

  const float4 qg = *(const float4*)(gs_rot + 4 * (size_t)n);
  const float rng = rnorm_clip(qg.x * qg.x + qg.y * qg.y + qg.z * qg.z + qg.w * qg.w);
  float Rg[9];
  quat_to_R(qg.x * rng, qg.y * rng, qg.z * rng, qg.w * rng, Rg);
  float Rw[9];
  mat3mul(Rr, Rg, Rw);

  // ---- neighbor loop ------------------------------------------------------
  const int* knnRow = knn + (size_t)a * KNN;
  float acc[FEAT];
#pragma unroll
  for (int f = 0; f < FEAT; ++f) acc[f] = 0.0f;
  float mu0 = 0.f, mu1 = 0.f, mu2 = 0.f;
  float qb0 = 0.f, qb1 = 0.f, qb2 = 0.f, qb3 = 0.f;
  float wsum = 0.f;

  int mNext = knnRow[0];
  for (int k = 0; k < KNN; ++k) {
    const int m = mNext;
    if (k + 1 < KNN) {
      mNext = knnRow[k + 1];
      // emits global_prefetch_b8 -- hide L2 gather latency of next neighbor
      __builtin_prefetch(node_sem + (size_t)mNext * FEAT, 0, 1);
      __builtin_prefetch(node_quat + 4 * (refBase + (size_t)mNext), 0, 1);
    }

    // live pose (pre-normalized pack or direct gather)
    float4 A, Bv;
    if (USE_PACK) {
      A  = pack[2 * (size_t)m];
      Bv = pack[2 * (size_t)m + 1];
    } else {
      const float* pl = node_xyz + 3 * (liveBase + (size_t)m);
      const float4 ql = *(const float4*)(node_quat + 4 * (liveBase + (size_t)m));
      const float rnl =
          rnorm_clip(ql.x * ql.x + ql.y * ql.y + ql.z * ql.z + ql.w * ql.w);
      A  = make_float4(pl[0], pl[1], pl[2], ql.x * rnl);
      Bv = make_float4(ql.y * rnl, ql.z * rnl, ql.w * rnl, 0.0f);
    }

    // RBF weight
    const float inv = s_inv[m];                       // LDS (async-staged)
    const float* pr = node_xyz + 3 * (refBase + (size_t)m);
    const float dx = xwx - pr[0], dy = xwy - pr[1], dz = xwz - pr[2];
    const float dsq = dx * dx + dy * dy + dz * dz;
    const float w = __expf(-dsq * inv);

    // q_rel = qn(q_live) * conj(qn(q_ref)) with a single v_rsq
    const float4 qr = *(const float4*)(node_quat + 4 * (refBase + (size_t)m));
    const float rn =
        rnorm_clip(qr.x * qr.x + qr.y * qr.y + qr.z * qr.z + qr.w * qr.w);
    const float bw = qr.x * rn, bx = qr.y * rn, by = qr.z * rn, bz = qr.w * rn;
    const float lw = A.w, lx = Bv.x, ly = Bv.y, lz = Bv.z;
    const float rw_ =  lw * bw + lx * bx + ly * by + lz * bz;
    const float rx_ = -lw * bx + lx * bw - ly * bz + lz * by;
    const float ry_ = -lw * by + lx * bz + ly * bw - lz * bx;
    const float rz_ = -lw * bz - lx * by + ly * bx + lz * bw;

    float Rl[9];
    quat_to_R(rw_, rx_, ry_, rz_, Rl);   // q_rel is unit: q2R renorm is identity
    const float mx = Rl[0] * dx + Rl[1] * dy + Rl[2] * dz + A.x;
    const float my = Rl[3] * dx + Rl[4] * dy + Rl[5] * dz + A.y;
    const float mz = Rl[6] * dx + Rl[7] * dy + Rl[8] * dz + A.z;

    mu0 = fmaf(w, mx, mu0);  mu1 = fmaf(w, my, mu1);  mu2 = fmaf(w, mz, mu2);
    qb0 = fmaf(w, rw_, qb0); qb1 = fmaf(w, rx_, qb1);
    qb2 = fmaf(w, ry_, qb2); qb3 = fmaf(w, rz_, qb3);
    wsum += w;

    const float4* srow = (const float4*)(node_sem + (size_t)m * FEAT);
#pragma unroll
    for (int j = 0; j < FEAT / 4; ++j) {
      const float4 v = srow[j];
      acc[4 * j + 0] = fmaf(w, v.x, acc[4 * j + 0]);
      acc[4 * j + 1] = fmaf(w, v.y, acc[4 * j + 1]);
      acc[4 * j + 2] = fmaf(w, v.z, acc[4 * j + 2]);
      acc[4 * j + 3] = fmaf(w, v.w, acc[4 * j + 3]);
    }
  }

  // ---- epilogue: normalize blends, build fr, misc activations -------------
  const float invW = fast_rcp(wsum + EPS);

  float* out_mu  = out;
  float* out_fr  = out + 3  * (size_t)N;
  float* out_s   = out + 12 * (size_t)N;
  float* out_o   = out + 15 * (size_t)N;
  float* out_sph = out + 16 * (size_t)N;
  float* out_sem = out + 28 * (size_t)N;

  out_mu[3 * (size_t)n + 0] = mu0 * invW;
  out_mu[3 * (size_t)n + 1] = mu1 * invW;
  out_mu[3 * (size_t)n + 2] = mu2 * invW;

  // fr = q2R(qnorm(q_blend)) @ R_world  (blend scale cancels inside q2R)
  const float rnb = rnorm_clip(qb0 * qb0 + qb1 * qb1 + qb2 * qb2 + qb3 * qb3);
  float Rb[9];
  quat_to_R(qb0 * rnb, qb1 * rnb, qb2 * rnb, qb3 * rnb, Rb);
  float Fr[9];
  mat3mul(Rb, Rw, Fr);
  float* fr = out_fr + 9 * (size_t)n;
#pragma unroll
  for (int i = 0; i < 9; ++i) fr[i] = Fr[i];

#pragma unroll
  for (int i = 0; i < 3; ++i)
    out_s[3 * (size_t)n + i] = MAX_S * fast_sigmoid(gs_scal[3 * (size_t)n + i]);
  out_o[n] = fast_sigmoid(gs_opa[n]);

  float* sp = out_sph + 12 * (size_t)n;
#pragma unroll
  for (int i = 0; i < 3; ++i) sp[i] = feat_dc[3 * (size_t)n + i];
#pragma unroll
  for (int i = 0; i < 9; ++i) sp[3 + i] = feat_rest[9 * (size_t)n + i];

  float4* so = (float4*)(out_sem + (size_t)n * FEAT);
#pragma unroll
  for (int j = 0; j < FEAT / 4; ++j)
    so[j] = make_float4(acc[4 * j + 0] * invW, acc[4 * j + 1] * invW,
                        acc[4 * j + 2] * invW, acc[4 * j + 3] * invW);
}

}  // namespace

// ---------------------------------------------------------------------------
extern "C" void kernel_launch(void* const* d_in, const int* in_sizes, int n_in,
                              void* d_out, int out_size, void* d_ws,
                              size_t ws_size, hipStream_t stream) {
  const float* gs_xyz     = (const float*)d_in[0];
  const float* gs_rot     = (const float*)d_in[1];
  const float* gs_scal    = (const float*)d_in[2];
  const float* gs_opa     = (const float*)d_in[3];
  const float* feat_dc    = (const float*)d_in[4];
  const float* feat_rest  = (const float*)d_in[5];
  const float* node_xyz   = (const float*)d_in[6];
  const float* node_quat  = (const float*)d_in[7];
  const float* node_sigma = (const float*)d_in[8];
  const float* node_sem   = (const float*)d_in[9];
  const int*   attach     = (const int*)d_in[10];
  const int*   reft       = (const int*)d_in[11];
  const int*   knn        = (const int*)d_in[12];
  const int*   tptr       = (const int*)d_in[13];

  const int N = in_sizes[0] / 3;   // gs_xyz is (N,3)
  const int M = in_sizes[8];       // node_sigma is (M,1)

  const size_t packBytes = (size_t)2 * (size_t)M * sizeof(float4);
  const bool usePack = (d_ws != nullptr) && (ws_size >= packBytes);

  float* out = (float*)d_out;
  const int gridN = (N + BLOCK - 1) / BLOCK;

  if (usePack) {
    float4* pack = (float4*)d_ws;
    prep_nodes_kernel<<<(M + BLOCK - 1) / BLOCK, BLOCK, 0, stream>>>(
        node_xyz, node_quat, tptr, pack, M);
    dynscf_kernel<true><<<gridN, BLOCK, 0, stream>>>(
        gs_xyz, gs_rot, gs_scal, gs_opa, feat_dc, feat_rest, node_xyz,
        node_quat, node_sigma, node_sem, attach, reft, knn, tptr,
        (const float4*)pack, out, N, M);
  } else {
    dynscf_kernel<false><<<gridN, BLOCK, 0, stream>>>(
        gs_xyz, gs_rot, gs_scal, gs_opa, feat_dc, feat_rest, node_xyz,
        node_quat, node_sigma, node_sem, attach, reft, knn, tptr,
        (const float4*)nullptr, out, N, M);
  }
}